// NeighborhoodAttentionAggregator_6923487281873
// MI455X (gfx1250) — compile-verified
//
#include <hip/hip_runtime.h>
#include <hip/hip_bf16.h>
#include <math.h>

// MI455X / gfx1250, wave32. Fused neighborhood-attention aggregator.
// Kernel 1: gate = sigmoid(center @ Wg), Hc = center @ Wc + b1  (WMMA f32 16x16x4)
// Kernel 2: per 2 centers: whole 64x256 neighbor tile DMA'd to LDS by the
//           Tensor Data Mover (hardware-padded rows), Wn staged via
//           async-to-LDS, WMMA score GEMM, exact GELU, @W2 reduction,
//           softmax, LDS-resident aggregation, gated residual.
//           ~141 KB LDS -> 2 blocks/WGP so staging overlaps compute.

typedef __attribute__((ext_vector_type(2))) float v2f;
typedef __attribute__((ext_vector_type(8))) float v8f;
typedef __attribute__((ext_vector_type(4))) int v4i;
typedef __attribute__((ext_vector_type(8))) int v8i;
typedef __attribute__((ext_vector_type(4))) unsigned v4u;

#define BB 16384
#define KK 32
#define DD 256
#define AA 64
#define TBC 2                      // centers per block in kernel 2
#define NEGV (-1000000000.0f)

// LDS strides (floats), padded for bank-conflict-free fragment reads
#define WS 72    // Wn rows (64 cols + pad): row offset 72 % 64 == 8
#define XS 260   // X / C rows (256 cols + pad): row offset 260 % 64 == 4
#define HS 68    // Hc rows

typedef __attribute__((address_space(1))) v4i g_v4i;   // global (AS1)
typedef __attribute__((address_space(3))) v4i l_v4i;   // LDS (AS3)
typedef __attribute__((address_space(3))) void l_void; // LDS (AS3)

#if __has_builtin(__builtin_amdgcn_global_load_async_to_lds_b128)
#define HAVE_ASYNC_LDS 1
__device__ __forceinline__ void async_copy16(const void* g, void* l) {
  __builtin_amdgcn_global_load_async_to_lds_b128((g_v4i*)(void*)g,
                                                 (l_v4i*)l, 0, 0);
}
__device__ __forceinline__ void async_wait0() {
#if __has_builtin(__builtin_amdgcn_s_wait_asynccnt)
  __builtin_amdgcn_s_wait_asynccnt(0);
#else
  asm volatile("s_wait_asynccnt 0" ::: "memory");
#endif
}
#else
#define HAVE_ASYNC_LDS 0
#endif

#if __has_builtin(__builtin_amdgcn_tensor_load_to_lds) && \
    __has_builtin(__builtin_amdgcn_s_wait_tensorcnt)
#define HAVE_TDM 1
#else
#define HAVE_TDM 0
#endif

__device__ __forceinline__ unsigned lds_offset_bytes(const void* p) {
  return (unsigned)(size_t)(l_void*)(void*)p;
}

__device__ __forceinline__ v8f wmma_f32x4(v2f a, v2f b, v8f c) {
  // D = A(16x4) * B(4x16) + C(16x16), fp32
  return __builtin_amdgcn_wmma_f32_16x16x4_f32(false, a, false, b, (short)0, c,
                                               false, false);
}

__device__ __forceinline__ float gelu_exact(float x) {
  return 0.5f * x * (1.0f + erff(x * 0.70710678118654752f));
}

// ---------------------------------------------------------------------------
// Kernel 1: per block = 16 centers. 256 threads = 8 waves.
//   gate[b,:] = sigmoid(center[b,:] @ Wg)      (16x256 = 16 N-tiles, 2/wave)
//   Hc[b,:]   = center[b,:] @ W1[:256] + b1    (16x64  = 4 N-tiles, waves 0-3)
// ---------------------------------------------------------------------------
__global__ __launch_bounds__(256)
void gate_hc_kernel(const float* __restrict__ center,
                    const float* __restrict__ W1,
                    const float* __restrict__ b1,
                    const float* __restrict__ Wg,
                    float* __restrict__ gateW,
                    float* __restrict__ HcW) {
  __shared__ __align__(16) float sC[16 * XS];

  const int tid = threadIdx.x;
  const int b0 = blockIdx.x * 16;

  // Stage center tile (16x256) into LDS
  const float4* cg = (const float4*)(center + (size_t)b0 * DD);
  for (int i = tid; i < 16 * 64; i += 256) {
    int row = i >> 6, c4 = i & 63;
#if HAVE_ASYNC_LDS
    async_copy16(&cg[i], &sC[row * XS + c4 * 4]);
#else
    *(float4*)&sC[row * XS + c4 * 4] = cg[i];
#endif
  }
#if HAVE_ASYNC_LDS
  async_wait0();
#endif
  __syncthreads();

  const int lane = tid & 31;
  const int wv   = tid >> 5;
  const int l16  = lane & 15;
  const int hi   = lane >> 4;
  const int roff = hi * 2;

  // ---- gate GEMM: wave wv owns N-tiles (2wv, 2wv+1) of Wg ----
  for (int nt = 0; nt < 2; ++nt) {
    const int n0 = (wv * 2 + nt) * 16;
    v8f acc = {};
    for (int kk = 0; kk < DD; kk += 4) {
      v2f a = *(const v2f*)&sC[l16 * XS + kk + roff];
      v2f bf;
      bf[0] = Wg[(kk + roff) * DD + n0 + l16];       // L2-resident (256 KB)
      bf[1] = Wg[(kk + 1 + roff) * DD + n0 + l16];
      acc = wmma_f32x4(a, bf, acc);
    }
#pragma unroll
    for (int r = 0; r < 8; ++r) {
      int m = r + hi * 8;
      float g = __builtin_amdgcn_rcpf(1.0f + __expf(-acc[r]));
      gateW[(size_t)(b0 + m) * DD + n0 + l16] = g;
    }
  }

  // ---- Hc GEMM: waves 0..3 each own one N-tile of Wc = W1[:256] ----
  if (wv < 4) {
    const int n0 = wv * 16;
    v8f acc = {};
    for (int kk = 0; kk < DD; kk += 4) {
      v2f a = *(const v2f*)&sC[l16 * XS + kk + roff];
      v2f bf;
      bf[0] = W1[(kk + roff) * AA + n0 + l16];
      bf[1] = W1[(kk + 1 + roff) * AA + n0 + l16];
      acc = wmma_f32x4(a, bf, acc);
    }
    float bb = b1[n0 + l16];
#pragma unroll
    for (int r = 0; r < 8; ++r) {
      int m = r + hi * 8;
      HcW[(size_t)(b0 + m) * AA + n0 + l16] = acc[r] + bb;
    }
  }
}

// ---------------------------------------------------------------------------
// Kernel 2: per block = 2 centers, all 32 neighbors resident in LDS (1 HBM
// pass over neighbor_emb). 128 threads = 4 waves; wave t owns M-tile t.
// LDS rows are b-major: m = b*32 + k  (so the TDM can write the whole tile as
// one linear, hardware-padded DMA: 256 dwords data + 4 dwords pad == XS).
// ---------------------------------------------------------------------------
__global__ __launch_bounds__(128)
void attn_agg_kernel(const float* __restrict__ center,
                     const float* __restrict__ nemb,
                     const float* __restrict__ nw,
                     const int* __restrict__ nmask,
                     const float* __restrict__ W1,
                     const float* __restrict__ W2v,
                     const float* __restrict__ b2,
                     const float* __restrict__ gateW,
                     const float* __restrict__ HcW,
                     float* __restrict__ out) {
  __shared__ __align__(16) float sWn[DD * WS];        // 72 KB: W1[256:512]
  __shared__ __align__(16) float sX[TBC * KK * XS];   // 65 KB: 2b x 32k x 256d
  __shared__ float sHc[TBC * HS];
  __shared__ float sW2[AA];
  __shared__ float sScore[TBC * 34];

  const int tid = threadIdx.x;  // 128
  const int b0 = blockIdx.x * TBC;
  const int lane = tid & 31;
  const int wv   = tid >> 5;

#if HAVE_TDM
  // ---- TDM: DMA the whole 2x(32x256) f32 neighbor tile into sX ----
  if (wv == 0) {
    unsigned long long ga =
        (unsigned long long)(size_t)(nemb + (size_t)b0 * KK * DD);
    unsigned lds = lds_offset_bytes(&sX[0]);
    // D# group 0: count=1, lds_addr, global_addr, type=2 ("image")
    v4u g0 = {1u, lds, (unsigned)ga,
              (unsigned)((ga >> 32) & 0x01FFFFFFu) | (2u << 30)};
    // D# group 1: data_size=4B(2), pad_enable, pad_interval=256dw(7),
    //             pad_amount=4dw(3); tensor_dim0=8192, tensor_dim1=2,
    //             tile_dim0=8192, tile_dim1=2, tensor_dim0_stride=8192
    v8i g1 = {(int)((2u << 16) | (1u << 20) | (7u << 22) | (3u << 25)),
              (int)(8192u << 16),            // tensor_dim0[15:0] @ bits 63:48
              (int)((8192u >> 16) | (2u << 16)),  // td0[31:16] | td1[15:0]
              (int)(8192u << 16),            // td1[31:16]=0 | tile_dim0
              (int)2,                        // tile_dim1=2, tile_dim2=0
              (int)8192,                     // td0_stride[31:0]
              0, 0};
    v4i gz = {0, 0, 0, 0};
    v8i gz8 = {0, 0, 0, 0, 0, 0, 0, 0};
    __builtin_amdgcn_tensor_load_to_lds(g0, g1, gz, gz, gz8, 0);
  }
#else
  for (int i = tid; i < TBC * 2048; i += 128) {       // 4096 float4
    int b = i >> 11;
    int rem = i & 2047;
    int k = rem >> 6, c4 = rem & 63;
    const float4* src = ((const float4*)(nemb + (size_t)(b0 + b) * KK * DD)) + rem;
    float* dst = &sX[(b * KK + k) * XS + c4 * 4];
#if HAVE_ASYNC_LDS
    async_copy16(src, dst);
#else
    *(float4*)dst = *src;
#endif
  }
#endif

  // Stage Wn (256x64) -> LDS (hits L2: 64 KB, shared by all blocks)
  const float4* wn4 = (const float4*)(W1 + DD * AA);
  for (int i = tid; i < DD * 16; i += 128) {          // 4096 float4
    int row = i >> 4, c4 = i & 15;
#if HAVE_ASYNC_LDS
    async_copy16(&wn4[i], &sWn[row * WS + c4 * 4]);
#else
    *(float4*)&sWn[row * WS + c4 * 4] = wn4[i];
#endif
  }
  if (tid < TBC * AA) {
    int b = tid >> 6, n = tid & 63;
    sHc[b * HS + n] = HcW[(size_t)(b0 + b) * AA + n];
  }
  if (tid < AA) sW2[tid] = W2v[tid];
#if HAVE_ASYNC_LDS
  async_wait0();
#endif
#if HAVE_TDM
  if (wv == 0) __builtin_amdgcn_s_wait_tensorcnt(0);
#endif
  __syncthreads();

  const int l16  = lane & 15;
  const int hi   = lane >> 4;
  const int roff = hi * 2;

  // ---- score GEMM: h = X @ Wn + Hc; tile t rows m=16t+rr, b=m>>5, k=m&31 ----
  v8f acc[4];
  {
    const int bsel = wv >> 1;   // b is constant within a tile (b-major rows)
#pragma unroll
    for (int nb = 0; nb < 4; ++nb) {
      float cinit = sHc[bsel * HS + nb * 16 + l16];
#pragma unroll
      for (int r = 0; r < 8; ++r) acc[nb][r] = cinit;
    }
  }
  const int arow = 16 * wv + l16;
  for (int kk = 0; kk < DD; kk += 4) {
    v2f a = *(const v2f*)&sX[arow * XS + kk + roff];
#pragma unroll
    for (int nb = 0; nb < 4; ++nb) {
      v2f bf;
      bf[0] = sWn[(kk + roff) * WS + nb * 16 + l16];
      bf[1] = sWn[(kk + 1 + roff) * WS + nb * 16 + l16];
      acc[nb] = wmma_f32x4(a, bf, acc[nb]);
    }
  }

  // ---- scores: gelu(h) @ W2, 16-lane tree reduction over the A dimension ----
  float part[8];
#pragma unroll
  for (int r = 0; r < 8; ++r) part[r] = 0.0f;
#pragma unroll
  for (int nb = 0; nb < 4; ++nb) {
    float w2 = sW2[nb * 16 + l16];
#pragma unroll
    for (int r = 0; r < 8; ++r) part[r] += gelu_exact(acc[nb][r]) * w2;
  }
#pragma unroll
  for (int r = 0; r < 8; ++r) {
    part[r] += __shfl_xor(part[r], 1, 16);
    part[r] += __shfl_xor(part[r], 2, 16);
    part[r] += __shfl_xor(part[r], 4, 16);
    part[r] += __shfl_xor(part[r], 8, 16);
  }
  if (l16 == 0) {
    float bias2 = b2[0];
#pragma unroll
    for (int r = 0; r < 8; ++r) {
      int rr = r + hi * 8;
      int m = 16 * wv + rr;
      int b = m >> 5;
      int k = m & 31;
      float sc = part[r] + bias2 + nw[(size_t)(b0 + b) * KK + k];
      bool ok = nmask[(size_t)(b0 + b) * KK + k] != 0;
      sScore[b * 34 + k] = ok ? sc : NEGV;
    }
  }
  __syncthreads();

  // ---- softmax over K=32: wave wv (<TBC) handles row b = wv, lane = k ----
  if (wv < TBC) {
    float s = sScore[wv * 34 + lane];
    float mx = s;
    mx = fmaxf(mx, __shfl_xor(mx, 16, 32));
    mx = fmaxf(mx, __shfl_xor(mx, 8, 32));
    mx = fmaxf(mx, __shfl_xor(mx, 4, 32));
    mx = fmaxf(mx, __shfl_xor(mx, 2, 32));
    mx = fmaxf(mx, __shfl_xor(mx, 1, 32));
    float e = __expf(s - mx);
    float sum = e;
    sum += __shfl_xor(sum, 16, 32);
    sum += __shfl_xor(sum, 8, 32);
    sum += __shfl_xor(sum, 4, 32);
    sum += __shfl_xor(sum, 2, 32);
    sum += __shfl_xor(sum, 1, 32);
    sScore[wv * 34 + lane] = e * __builtin_amdgcn_rcpf(sum);
  }
  __syncthreads();

  // ---- aggregation (from LDS tile — no second HBM read) + gated residual ----
  for (int idx = tid; idx < TBC * DD; idx += 128) {
    int b = idx >> 8;
    int d = idx & 255;
    float agg = 0.0f;
#pragma unroll 8
    for (int k = 0; k < KK; ++k)
      agg = fmaf(sScore[b * 34 + k], sX[(b * KK + k) * XS + d], agg);
    size_t o = (size_t)(b0 + b) * DD + d;
    out[o] = center[o] + gateW[o] * agg;
  }
}

// ---------------------------------------------------------------------------
extern "C" void kernel_launch(void* const* d_in, const int* in_sizes, int n_in,
                              void* d_out, int out_size, void* d_ws, size_t ws_size,
                              hipStream_t stream) {
  const float* center = (const float*)d_in[0];   // (B, D)
  const float* nemb   = (const float*)d_in[1];   // (B, K, D)
  const float* nw     = (const float*)d_in[2];   // (B, K)
  const int*   nmask  = (const int*)d_in[3];     // (B, K) bool->int
  const float* W1     = (const float*)d_in[4];   // (2D, A)
  const float* b1     = (const float*)d_in[5];   // (A,)
  const float* W2v    = (const float*)d_in[6];   // (A, 1)
  const float* b2     = (const float*)d_in[7];   // (1,)
  const float* Wg     = (const float*)d_in[8];   // (D, D)
  float* out = (float*)d_out;

  float* gateW = (float*)d_ws;                   // B*D floats (16 MB)
  float* HcW   = gateW + (size_t)BB * DD;        // B*A floats (4 MB)

  gate_hc_kernel<<<BB / 16, 256, 0, stream>>>(center, W1, b1, Wg, gateW, HcW);
  attn_agg_kernel<<<BB / TBC, 128, 0, stream>>>(center, nemb, nw, nmask, W1,
                                                W2v, b2, gateW, HcW, out);
}